// L1Attn_36859409334845
// MI455X (gfx1250) — compile-verified
//
#include <hip/hip_runtime.h>
#include <cstdint>

// L1 attention scores on gfx1250 (MI455X).
// out[b][h][j][i] = -(1/8) * sum_d |q[b,i,h,d] - k[b,j,h,d]|
// Shapes fixed by the harness: B=2, S=512, H=8, D=64.
//
// VALU-bound problem (|q-k| is not bilinear -> WMMA inapplicable exactly).
// CDNA5-specific paths used:
//   * async global->LDS copy engine (global_load_async_to_lds_b128,
//     s_wait_asynccnt, ASYNCcnt-tracked)
//   * wave32 register blocking, VOPD-friendly VALU core
//   * non-temporal output stores (write-once 16 MB stream)

typedef float v4f __attribute__((ext_vector_type(4)));  // native vector for b128/NT

namespace {
constexpr int B = 2;
constexpr int S = 512;
constexpr int H = 8;
constexpr int D = 64;

constexpr int TILE = 64;        // 64 q-rows x 64 k-rows per block
constexpr int LDS_STRIDE = 68;  // floats; 68*4 = 272 B = 17*16 -> 16B-aligned rows,
                                // odd multiple of 16B breaks bank-conflict periodicity
}  // namespace

__global__ __launch_bounds__(256)
void l1attn_gfx1250_kernel(const float* __restrict__ q,
                           const float* __restrict__ k,
                           float* __restrict__ out) {
  __shared__ alignas(16) float QT[TILE * LDS_STRIDE];  // [i_local][d], padded
  __shared__ alignas(16) float KT[TILE * LDS_STRIDE];  // [j_local][d], padded

  const int bh = blockIdx.z;           // 0..15
  const int b  = bh >> 3;
  const int h  = bh & 7;
  const int i0 = blockIdx.x * TILE;    // q-position tile
  const int j0 = blockIdx.y * TILE;    // k-position tile

  // Row r of the tile lives at base + r * (H*D) floats (row stride 2048 B).
  const float* qbase = q + ((size_t)(b * S + i0) * H + h) * D;
  const float* kbase = k + ((size_t)(b * S + j0) * H + h) * D;

  const int t = threadIdx.x;

  // ---- Stage q/k tiles into LDS with the CDNA5 async copy engine ----------
  // 64 rows x 64 floats = 1024 x 16B chunks per matrix; 256 threads x 4 chunks.
  const uint32_t qt_base = (uint32_t)(uintptr_t)(void*)QT;  // low 32 bits = LDS offset
  const uint32_t kt_base = (uint32_t)(uintptr_t)(void*)KT;
#pragma unroll
  for (int it = 0; it < 4; ++it) {
    const int c   = t + it * 256;          // chunk id 0..1023
    const int row = c >> 4;                // 0..63
    const int col = c & 15;                // 0..15 (16B units within a row)
    const uint32_t goff = (uint32_t)(row * (H * D * 4) + col * 16);
    const uint32_t loff = (uint32_t)(row * (LDS_STRIDE * 4) + col * 16);
    const uint32_t qdst = qt_base + loff;
    const uint32_t kdst = kt_base + loff;
    // GVS mode: 64-bit SGPR base + 32-bit per-lane VGPR offset, copied to the LDS address in vdst.
    asm volatile("global_load_async_to_lds_b128 %0, %1, %2"
                 :: "v"(qdst), "v"(goff), "s"(qbase) : "memory");
    asm volatile("global_load_async_to_lds_b128 %0, %1, %2"
                 :: "v"(kdst), "v"(goff), "s"(kbase) : "memory");
  }
  asm volatile("s_wait_asynccnt 0" ::: "memory");  // this wave's copies done
  __syncthreads();                                 // all waves' copies visible

  // ---- Compute: each thread owns a 4(j) x 4(i) micro-tile ------------------
  const int ti = t & 15;   // i sub-tile
  const int tj = t >> 4;   // j sub-tile
  const float* qrow = &QT[(ti * 4) * LDS_STRIDE];
  const float* krow = &KT[(tj * 4) * LDS_STRIDE];

  float acc[4][4] = {{0.f}};

#pragma unroll 4
  for (int dc = 0; dc < D / 4; ++dc) {   // d in chunks of 4 (v4f LDS reads)
    v4f qv[4], kv[4];
#pragma unroll
    for (int r = 0; r < 4; ++r) {
      qv[r] = *(const v4f*)(qrow + r * LDS_STRIDE + dc * 4);
      kv[r] = *(const v4f*)(krow + r * LDS_STRIDE + dc * 4);
    }
#pragma unroll
    for (int jj = 0; jj < 4; ++jj) {
#pragma unroll
      for (int ii = 0; ii < 4; ++ii) {
        // abs folds into a VALU source modifier: v_sub_f32 + v_add_f32 |x|
        acc[jj][ii] += fabsf(qv[ii].x - kv[jj].x);
        acc[jj][ii] += fabsf(qv[ii].y - kv[jj].y);
        acc[jj][ii] += fabsf(qv[ii].z - kv[jj].z);
        acc[jj][ii] += fabsf(qv[ii].w - kv[jj].w);
      }
    }
  }

  // ---- Store: out[((bh*S + j)*S) + i], i contiguous -> NT v4f stores -------
  // Output is a write-once 16 MB stream: non-temporal hint keeps it from
  // evicting the q/k working set (which IS reused across 8 blocks per row).
  const float scale = -0.125f;  // -1/sqrt(64)
  float* obase = out + ((size_t)bh * S + (size_t)(j0 + tj * 4)) * S
                     + (size_t)(i0 + ti * 4);
#pragma unroll
  for (int jj = 0; jj < 4; ++jj) {
    v4f o;
    o.x = scale * acc[jj][0];
    o.y = scale * acc[jj][1];
    o.z = scale * acc[jj][2];
    o.w = scale * acc[jj][3];
    __builtin_nontemporal_store(o, (v4f*)(obase + (size_t)jj * S));
  }
}

extern "C" void kernel_launch(void* const* d_in, const int* in_sizes, int n_in,
                              void* d_out, int out_size, void* d_ws, size_t ws_size,
                              hipStream_t stream) {
  (void)in_sizes; (void)n_in; (void)out_size; (void)d_ws; (void)ws_size;
  const float* q = (const float*)d_in[0];
  const float* k = (const float*)d_in[1];
  float* out = (float*)d_out;

  dim3 grid(S / TILE, S / TILE, B * H);  // (8, 8, 16)
  dim3 block(256, 1, 1);                 // 8 wave32 waves
  l1attn_gfx1250_kernel<<<grid, block, 0, stream>>>(q, k, out);
}